// GlobalAttention_72559177499312
// MI455X (gfx1250) — compile-verified
//
#include <hip/hip_runtime.h>

#define B_ 64
#define S_ 4096
#define D_ 1024
#define NCHUNK 16
#define CROWS (S_ / NCHUNK) /* 256 rows per chunk */

typedef __attribute__((ext_vector_type(2))) float v2f;
typedef __attribute__((ext_vector_type(8))) float v8f;

// ---------------------------------------------------------------------------
// Kernel 1: C[m][n] = sum_k A[m][k] * W[n][k]   (C = A @ W^T)
// A: [64 x 1024], W: [1024 x 1024] row-major. One 16x16 tile per wave,
// K-loop over V_WMMA_F32_16X16X4_F32.
// A-frag layout (16x4 f32): lanes 0-15 hold K=k,k+1 for M=lane; lanes 16-31
// hold K=k+2,k+3. B-frag symmetric with N=lane. C/D: VGPR r -> M=r (+8 for
// upper half-lanes), N = lane%16.
// ---------------------------------------------------------------------------
__global__ __launch_bounds__(128) void wmma_gemm_at(const float* __restrict__ A,
                                                    const float* __restrict__ W,
                                                    float* __restrict__ C) {
  const int wave = threadIdx.x >> 5;
  const int lane = threadIdx.x & 31;
  const int tile = blockIdx.x * 4 + wave;   // 256 tiles total
  const int m0 = (tile & 3) * 16;           // M = 64 -> 4 m-tiles
  const int n0 = (tile >> 2) * 16;          // N = 1024 -> 64 n-tiles
  const int half = lane >> 4;
  const int lid = lane & 15;
  const float* arow = A + (size_t)(m0 + lid) * D_ + 2 * half;
  const float* brow = W + (size_t)(n0 + lid) * D_ + 2 * half;
  v8f c = {0.f, 0.f, 0.f, 0.f, 0.f, 0.f, 0.f, 0.f};
#pragma unroll 8
  for (int k = 0; k < D_; k += 4) {
    v2f a = *(const v2f*)(arow + k);
    v2f b = *(const v2f*)(brow + k);
    c = __builtin_amdgcn_wmma_f32_16x16x4_f32(false, a, false, b, (short)0, c,
                                              false, false);
  }
  float* crow = C + (size_t)(m0 + 8 * half) * D_ + n0 + lid;
#pragma unroll
  for (int r = 0; r < 8; ++r) crow[(size_t)r * D_] = c[r];
}

// ---------------------------------------------------------------------------
// Kernel 2: masked flash pass over one chunk of 256 context rows.
// One wave32 per workgroup. Each lane owns d = 4*lane + 128*i (i=0..7) of the
// 1024-wide row/accumulator -> 8 float4 loads of 512 contiguous bytes/wave.
// Masked rows (~50%) are skipped entirely (their context row is never read).
// Raw masked scores are written to the attention output region (normalized
// later); per-chunk (m, l, acc[1024]) go to workspace.
// ---------------------------------------------------------------------------
__global__ __launch_bounds__(32) void attn_chunk(
    const float* __restrict__ ctx, const float* __restrict__ q,
    const int* __restrict__ mask, float* __restrict__ raw_scores,
    float* __restrict__ chunk_acc, float* __restrict__ chunk_ml) {
  const int b = blockIdx.y;
  const int chunk = blockIdx.x;
  const int lane = threadIdx.x;
  const int s0 = chunk * CROWS;

  const float* qb = q + (size_t)b * D_;
  float4 qv[8];
#pragma unroll
  for (int i = 0; i < 8; ++i)
    qv[i] = *(const float4*)(qb + i * 128 + lane * 4);

  float4 acc[8];
#pragma unroll
  for (int i = 0; i < 8; ++i) acc[i] = make_float4(0.f, 0.f, 0.f, 0.f);
  float m = -INFINITY;
  float l = 0.f;

  const float* cbase = ctx + ((size_t)b * S_ + s0) * D_;

  for (int sb = 0; sb < CROWS; sb += 32) {
    // coalesced mask load for the next 32 rows; lane t holds mask of row sb+t
    const int mv = mask[(size_t)b * S_ + s0 + sb + lane];
    float myscore = -INFINITY;

    for (int t = 0; t < 32; ++t) {
      const int mk = __shfl(mv, t, 32);
      float score = -INFINITY;

      // prefetch next unmasked row while working on this one
      if (t < 31) {
        const int mkn = __shfl(mv, t + 1, 32);
        if (mkn != 0)
          __builtin_prefetch(cbase + (size_t)(sb + t + 1) * D_ + lane * 32, 0, 0);
      }

      if (mk != 0) {
        const float* row = cbase + (size_t)(sb + t) * D_;
        float4 rv[8];
#pragma unroll
        for (int i = 0; i < 8; ++i)
          rv[i] = *(const float4*)(row + i * 128 + lane * 4);

        float p0 = 0.f, p1 = 0.f, p2 = 0.f, p3 = 0.f;
#pragma unroll
        for (int i = 0; i < 8; ++i) {
          p0 = fmaf(rv[i].x, qv[i].x, p0);
          p1 = fmaf(rv[i].y, qv[i].y, p1);
          p2 = fmaf(rv[i].z, qv[i].z, p2);
          p3 = fmaf(rv[i].w, qv[i].w, p3);
        }
        float p = (p0 + p1) + (p2 + p3);
#pragma unroll
        for (int off = 16; off > 0; off >>= 1) p += __shfl_xor(p, off, 32);
        score = p;  // uniform across lanes

        if (score > m) {  // wave-uniform branch; rare after warmup
          const float sc = __expf(m - score);
          l *= sc;
#pragma unroll
          for (int i = 0; i < 8; ++i) {
            acc[i].x *= sc; acc[i].y *= sc; acc[i].z *= sc; acc[i].w *= sc;
          }
          m = score;
        }
        const float w = __expf(score - m);
        l += w;
#pragma unroll
        for (int i = 0; i < 8; ++i) {
          acc[i].x = fmaf(w, rv[i].x, acc[i].x);
          acc[i].y = fmaf(w, rv[i].y, acc[i].y);
          acc[i].z = fmaf(w, rv[i].z, acc[i].z);
          acc[i].w = fmaf(w, rv[i].w, acc[i].w);
        }
      }
      if (t == lane) myscore = score;
    }
    // coalesced raw-score store (32 rows at a time)
    raw_scores[(size_t)b * S_ + s0 + sb + lane] = myscore;
  }

  float* oa = chunk_acc + (size_t)(b * NCHUNK + chunk) * D_;
#pragma unroll
  for (int i = 0; i < 8; ++i)
    *(float4*)(oa + i * 128 + lane * 4) = acc[i];
  if (lane == 0) {
    chunk_ml[(b * NCHUNK + chunk) * 2 + 0] = m;
    chunk_ml[(b * NCHUNK + chunk) * 2 + 1] = l;
  }
}

// ---------------------------------------------------------------------------
// Kernel 3: combine per-chunk partials -> weight_context[b, :], final (M, 1/L)
// ---------------------------------------------------------------------------
__global__ __launch_bounds__(256) void attn_combine(
    const float* __restrict__ chunk_acc, const float* __restrict__ chunk_ml,
    float* __restrict__ wc_out, float* __restrict__ ml_final) {
  const int b = blockIdx.x;
  const int tid = threadIdx.x;
  __shared__ float sm[NCHUNK];
  __shared__ float sl[NCHUNK];
  if (tid < NCHUNK) {
    sm[tid] = chunk_ml[(b * NCHUNK + tid) * 2 + 0];
    sl[tid] = chunk_ml[(b * NCHUNK + tid) * 2 + 1];
  }
  __syncthreads();
  float M = -INFINITY;
#pragma unroll
  for (int j = 0; j < NCHUNK; ++j) M = fmaxf(M, sm[j]);
  float wexp[NCHUNK];
  float L = 0.f;
#pragma unroll
  for (int j = 0; j < NCHUNK; ++j) {
    wexp[j] = __expf(sm[j] - M);
    L += sl[j] * wexp[j];
  }
  const float invL = 1.0f / L;
#pragma unroll
  for (int r = 0; r < 4; ++r) {
    const int d = tid + r * 256;
    float v = 0.f;
#pragma unroll
    for (int j = 0; j < NCHUNK; ++j)
      v = fmaf(wexp[j], chunk_acc[(size_t)(b * NCHUNK + j) * D_ + d], v);
    wc_out[(size_t)b * D_ + d] = v * invL;
  }
  if (tid == 0) {
    ml_final[b * 2 + 0] = M;
    ml_final[b * 2 + 1] = invL;
  }
}

// ---------------------------------------------------------------------------
// Kernel 4: normalize raw scores in-place -> attention = exp(s - M) / L
// (exp(-inf - M) == 0 handles masked rows)
// ---------------------------------------------------------------------------
__global__ __launch_bounds__(256) void attn_norm(float* __restrict__ att,
                                                 const float* __restrict__ ml_final) {
  const int idx = blockIdx.x * 256 + threadIdx.x;  // B_*S_ total
  const int b = idx >> 12;                         // / S_
  const float M = ml_final[b * 2 + 0];
  const float invL = ml_final[b * 2 + 1];
  att[idx] = __expf(att[idx] - M) * invL;
}

// ---------------------------------------------------------------------------
// Kernel 5: out = tanh(wc @ W_context^T + query @ W_query^T), WMMA f32.
// ---------------------------------------------------------------------------
__global__ __launch_bounds__(128) void wmma_final(
    const float* __restrict__ WCin, const float* __restrict__ Wc,
    const float* __restrict__ Q, const float* __restrict__ Wq,
    float* __restrict__ OUT) {
  const int wave = threadIdx.x >> 5;
  const int lane = threadIdx.x & 31;
  const int tile = blockIdx.x * 4 + wave;
  const int m0 = (tile & 3) * 16;
  const int n0 = (tile >> 2) * 16;
  const int half = lane >> 4;
  const int lid = lane & 15;
  const size_t aoff = (size_t)(m0 + lid) * D_ + 2 * half;
  const size_t boff = (size_t)(n0 + lid) * D_ + 2 * half;
  v8f c = {0.f, 0.f, 0.f, 0.f, 0.f, 0.f, 0.f, 0.f};
#pragma unroll 8
  for (int k = 0; k < D_; k += 4) {
    v2f a = *(const v2f*)(WCin + aoff + k);
    v2f b = *(const v2f*)(Wc + boff + k);
    c = __builtin_amdgcn_wmma_f32_16x16x4_f32(false, a, false, b, (short)0, c,
                                              false, false);
  }
#pragma unroll 8
  for (int k = 0; k < D_; k += 4) {
    v2f a = *(const v2f*)(Q + aoff + k);
    v2f b = *(const v2f*)(Wq + boff + k);
    c = __builtin_amdgcn_wmma_f32_16x16x4_f32(false, a, false, b, (short)0, c,
                                              false, false);
  }
  float* crow = OUT + (size_t)(m0 + 8 * half) * D_ + n0 + lid;
#pragma unroll
  for (int r = 0; r < 8; ++r) crow[(size_t)r * D_] = tanhf(c[r]);
}

// ---------------------------------------------------------------------------
extern "C" void kernel_launch(void* const* d_in, const int* in_sizes, int n_in,
                              void* d_out, int out_size, void* d_ws,
                              size_t ws_size, hipStream_t stream) {
  (void)in_sizes; (void)n_in; (void)out_size; (void)ws_size;
  const float* query   = (const float*)d_in[0];  // [64,1024]
  const float* context = (const float*)d_in[1];  // [64,4096,1024]
  const int*   cmask   = (const int*)  d_in[2];  // [64,4096]
  const float* W_align = (const float*)d_in[3];  // [1024,1024]
  const float* W_ctx   = (const float*)d_in[4];  // [1024,1024]
  const float* W_qry   = (const float*)d_in[5];  // [1024,1024]

  // Output layout: out [64*1024] | attention [64*4096] | weight_context [64*1024]
  float* out_p = (float*)d_out;
  float* att_p = out_p + (size_t)B_ * D_;
  float* wc_p  = att_p + (size_t)B_ * S_;

  // Workspace layout (floats): q[64*1024] | chunk_acc[64*16*1024] | chunk_ml | ml_final
  float* q_ws      = (float*)d_ws;
  float* chunk_acc = q_ws + (size_t)B_ * D_;
  float* chunk_ml  = chunk_acc + (size_t)B_ * NCHUNK * D_;
  float* ml_final  = chunk_ml + (size_t)B_ * NCHUNK * 2;

  // 1) q = query @ W_align^T
  wmma_gemm_at<<<64, 128, 0, stream>>>(query, W_align, q_ws);

  // 2) masked single-pass flash over context (raw scores -> attention region)
  attn_chunk<<<dim3(NCHUNK, B_), 32, 0, stream>>>(context, q_ws, cmask, att_p,
                                                  chunk_acc, chunk_ml);

  // 3) combine chunk partials -> weight_context, final (M, 1/L)
  attn_combine<<<B_, 256, 0, stream>>>(chunk_acc, chunk_ml, wc_p, ml_final);

  // 4) normalize attention in place
  attn_norm<<<(B_ * S_) / 256, 256, 0, stream>>>(att_p, ml_final);

  // 5) out = tanh(wc @ W_context^T + query @ W_query^T)
  wmma_final<<<64, 128, 0, stream>>>(wc_p, W_ctx, query, W_qry, out_p);
}